// HGCNDecoder_5789615915457
// MI455X (gfx1250) — compile-verified
//
#include <hip/hip_runtime.h>
#include <math.h>

typedef __attribute__((ext_vector_type(2))) float v2f;
typedef __attribute__((ext_vector_type(8))) float v8f;
typedef __attribute__((ext_vector_type(4))) unsigned int u32x4;
typedef __attribute__((ext_vector_type(4))) int i32x4;
typedef __attribute__((ext_vector_type(8))) int i32x8;

#define DHID 128
#define MAXZ 16
#define EPSF 1e-15f
#define BNDF (1.0f - 1e-5f)
#define LDS_STRIDE 132  // 128 + 4 dword pad (matches TDM pad_interval=128dw, pad_amount=4dw)

// ---------------- wave32 reduction ----------------
__device__ __forceinline__ float wave_allsum(float v) {
#pragma unroll
  for (int off = 16; off > 0; off >>= 1)
    v += __shfl_xor(v, off, 32);
  return v;
}

// ---------------- WMMA wrapper (f32 16x16x4, full precision) ----------------
__device__ __forceinline__ v8f wmma_k4(v2f a, v2f b, v8f c) {
#if __has_builtin(__builtin_amdgcn_wmma_f32_16x16x4_f32)
  // 8 args: (neg_a, A, neg_b, B, c_mod, C, reuse_a, reuse_b)
  return __builtin_amdgcn_wmma_f32_16x16x4_f32(false, a, false, b, (short)0, c,
                                               false, false);
#else
  c[0] += a.x * b.x;  // placeholder; histogram would show wmma==0
  return c;
#endif
}

// ------------- TDM: DMA a 16x128 f32 tile (row-major) into padded LDS -------
// D# group0: count=1 | lds_addr | global_addr | type=2
// D# group1: data_size=4B, pad_enable (every 128 dwords insert 4 dwords),
//            tensor_dim0=128, tensor_dim1=rows_remaining (OOB rows read 0),
//            tile_dim0=128, tile_dim1=16, tensor_dim0_stride=128
__device__ __forceinline__ void tdm_load_tile(const float* gsrc,
                                              unsigned lds_addr,
                                              int rows_remaining) {
#if __has_builtin(__builtin_amdgcn_tensor_load_to_lds)
  unsigned long long ga = (unsigned long long)(uintptr_t)gsrc;
  u32x4 g0;
  g0.x = 1u;                                            // count=1, user mode
  g0.y = lds_addr;                                      // LDS byte address
  g0.z = (unsigned)ga;                                  // global_addr[31:0]
  g0.w = (unsigned)((ga >> 32) & 0x01FFFFFFu) | (2u << 30);  // [56:32] | type=2
  i32x8 g1;
  g1[0] = (2 << 16)            // data_size = 4 bytes
        | (1 << 20)            // pad_enable
        | (6 << 22)            // pad_interval: 2<<6 = 128 dwords
        | (3 << 25);           // pad_amount: 4 dwords -> LDS row stride 132
  g1[1] = (DHID & 0xFFFF) << 16;                        // tensor_dim0 lo16
  g1[2] = ((unsigned)DHID >> 16)                        // tensor_dim0 hi16 (0)
        | ((rows_remaining & 0xFFFF) << 16);            // tensor_dim1 lo16
  g1[3] = ((rows_remaining >> 16) & 0xFFFF)             // tensor_dim1 hi16
        | (DHID << 16);                                 // tile_dim0 = 128
  g1[4] = 16;                                           // tile_dim1=16, dim2=0
  g1[5] = DHID;                                         // dim0_stride lo32
  g1[6] = 0;                                            // dim0_stride hi16, dim1_stride lo16
  g1[7] = 0;                                            // dim1_stride hi32
  i32x4 z4 = {0, 0, 0, 0};
#if __clang_major__ >= 23
  i32x8 z8 = {0, 0, 0, 0, 0, 0, 0, 0};
  __builtin_amdgcn_tensor_load_to_lds(g0, g1, z4, z4, z8, 0);
#else
  __builtin_amdgcn_tensor_load_to_lds(g0, g1, z4, z4, 0);
#endif
  __builtin_amdgcn_s_wait_tensorcnt(0);
#endif
}

// ---------------- row-norm elementwise kernels (1 wave per 128-wide row) ----
__global__ void k_expmap0(const float* __restrict__ in, float* __restrict__ out,
                          int nrows) {
  int row  = blockIdx.x * 8 + (threadIdx.x >> 5);
  int lane = threadIdx.x & 31;
  if (row >= nrows) return;
  float4 v = ((const float4*)(in + (size_t)row * DHID))[lane];
  float ss = wave_allsum(v.x * v.x + v.y * v.y + v.z * v.z + v.w * v.w);
  float n = fmaxf(sqrtf(ss), EPSF);
  float s = tanhf(n) / n;
  v.x *= s; v.y *= s; v.z *= s; v.w *= s;
  ((float4*)(out + (size_t)row * DHID))[lane] = v;
}

__global__ void k_logmap0(const float* __restrict__ in, float* __restrict__ out,
                          int nrows) {
  int row  = blockIdx.x * 8 + (threadIdx.x >> 5);
  int lane = threadIdx.x & 31;
  if (row >= nrows) return;
  float4 v = ((const float4*)(in + (size_t)row * DHID))[lane];
  float ss = wave_allsum(v.x * v.x + v.y * v.y + v.z * v.z + v.w * v.w);
  float n = fmaxf(sqrtf(ss), EPSF);
  float a = fminf(n, BNDF);
  float s = (0.5f * logf((1.0f + a) / (1.0f - a))) / n;  // artanh(a)/n
  v.x *= s; v.y *= s; v.z *= s; v.w *= s;
  ((float4*)(out + (size_t)row * DHID))[lane] = v;
}

__global__ void k_relu_expmap(const float* __restrict__ xlin,
                              const float* __restrict__ agg,
                              const float* __restrict__ node_mask,
                              float* __restrict__ out, int nrows) {
  int row  = blockIdx.x * 8 + (threadIdx.x >> 5);
  int lane = threadIdx.x & 31;
  if (row >= nrows) return;
  float m = node_mask[row];
  float4 a = ((const float4*)(xlin + (size_t)row * DHID))[lane];
  float4 g = ((const float4*)(agg + (size_t)row * DHID))[lane];
  float4 v;
  v.x = fmaxf(a.x + g.x, 0.0f) * m;
  v.y = fmaxf(a.y + g.y, 0.0f) * m;
  v.z = fmaxf(a.z + g.z, 0.0f) * m;
  v.w = fmaxf(a.w + g.w, 0.0f) * m;
  float ss = wave_allsum(v.x * v.x + v.y * v.y + v.z * v.z + v.w * v.w);
  float n = fmaxf(sqrtf(ss), EPSF);
  float s = tanhf(n) / n;
  v.x *= s; v.y *= s; v.z *= s; v.w *= s;
  ((float4*)(out + (size_t)row * DHID))[lane] = v;
}

// ---------------- 128x128 GEMM via f32 WMMA + TDM-staged A tile -------------
__global__ void k_gemm128(const float* __restrict__ X, const float* __restrict__ W,
                          const float* __restrict__ bias, float* __restrict__ Y,
                          int nrows) {
  __shared__ float Alds[16 * LDS_STRIDE];
  int rowBase = blockIdx.x * 16;
  int wave = threadIdx.x >> 5;
  int lane = threadIdx.x & 31;

#if __has_builtin(__builtin_amdgcn_tensor_load_to_lds)
  if (wave == 0) {
    tdm_load_tile(X + (size_t)rowBase * DHID,
                  (unsigned)(uintptr_t)(void*)Alds,  // low 32 bits = LDS offset
                  nrows - rowBase);
  }
  __syncthreads();
#else
  for (int i = threadIdx.x; i < 16 * DHID; i += 256) {
    int r = i >> 7, c = i & 127;
    float v = 0.0f;
    if (rowBase + r < nrows) v = X[(size_t)(rowBase + r) * DHID + c];
    Alds[r * LDS_STRIDE + c] = v;
  }
  __syncthreads();
#endif

  int n0 = wave * 16;
  int m  = lane & 15;          // A row for this lane / B,C column
  int kk = (lane >> 4) * 2;    // lanes 16-31 hold K+2,K+3 (ISA 16x4 A layout)
  int n  = n0 + m;

  v8f c = {};
#pragma unroll 4
  for (int k0 = 0; k0 < DHID; k0 += 4) {
    v2f a, b;
    a.x = Alds[m * LDS_STRIDE + k0 + kk];
    a.y = Alds[m * LDS_STRIDE + k0 + kk + 1];
    b.x = W[(size_t)(k0 + kk) * DHID + n];
    b.y = W[(size_t)(k0 + kk + 1) * DHID + n];
    c = wmma_k4(a, b, c);
  }

  float bn = bias[n];
  int rsel = (lane >> 4) * 8;  // C/D layout: lanes 16-31 hold M = v+8
  if (rowBase + 16 <= nrows) { // uniform fast path: full tile, no lane guards
#pragma unroll
    for (int v = 0; v < 8; ++v)
      Y[(size_t)(rowBase + rsel + v) * DHID + n] = c[v] + bn;
  } else {
#pragma unroll
    for (int v = 0; v < 8; ++v) {
      int row = rowBase + rsel + v;
      if (row < nrows) Y[(size_t)row * DHID + n] = c[v] + bn;
    }
  }
}

// ---------------- output GEMM: [N,128] @ [128,16] + bout --------------------
__global__ void k_gemm_out(const float* __restrict__ X,
                           const float* __restrict__ Wout,
                           const float* __restrict__ bout,
                           float* __restrict__ Y, int nrows) {
  int tile = blockIdx.x * 8 + (threadIdx.x >> 5);
  if (tile * 16 >= nrows) return;  // wave-uniform guard, EXEC stays all-ones
  int lane = threadIdx.x & 31;
  int rowBase = tile * 16;
  int m  = lane & 15;
  int kk = (lane >> 4) * 2;

  v8f c = {};
#pragma unroll 4
  for (int k0 = 0; k0 < DHID; k0 += 4) {
    v2f a, b;
    a.x = X[(size_t)(rowBase + m) * DHID + k0 + kk];
    a.y = X[(size_t)(rowBase + m) * DHID + k0 + kk + 1];
    b.x = Wout[(size_t)(k0 + kk) * MAXZ + m];
    b.y = Wout[(size_t)(k0 + kk + 1) * MAXZ + m];
    c = wmma_k4(a, b, c);
  }

  float bn = bout[m];
  int rsel = (lane >> 4) * 8;
#pragma unroll
  for (int v = 0; v < 8; ++v)
    Y[(size_t)(rowBase + rsel + v) * MAXZ + m] = c[v] + bn;
}

// ---------------- aggregation -----------------------------------------------
__global__ void k_zero(float4* __restrict__ p, long long n4) {
  long long i = blockIdx.x * 256LL + threadIdx.x;
  if (i < n4) p[i] = make_float4(0.f, 0.f, 0.f, 0.f);
}

// edge-parallel scatter: agg[row] += xlin[col] * edge_mask[e]
// one thread per (edge, 4-float chunk); targets are L2-resident -> fast atomics
__global__ void k_scatter(const float* __restrict__ xlin,
                          const int* __restrict__ edges,
                          const float* __restrict__ edge_mask,
                          float* __restrict__ agg, int nedges) {
  long long tid = blockIdx.x * 256LL + threadIdx.x;
  if (tid >= (long long)nedges * 32) return;
  int e  = (int)(tid >> 5);
  int c4 = (int)(tid & 31);
  int row = edges[e];            // edges[0][e]
  int col = edges[nedges + e];   // edges[1][e]
  float m = edge_mask[e];
  float4 v = ((const float4*)(xlin + (size_t)col * DHID))[c4];
  float* dst = agg + (size_t)row * DHID + c4 * 4;
  unsafeAtomicAdd(dst + 0, v.x * m);   // global_atomic_add_f32, no return
  unsafeAtomicAdd(dst + 1, v.y * m);
  unsafeAtomicAdd(dst + 2, v.z * m);
  unsafeAtomicAdd(dst + 3, v.w * m);
}

// ---------------- launch ----------------------------------------------------
extern "C" void kernel_launch(void* const* d_in, const int* in_sizes, int n_in,
                              void* d_out, int out_size, void* d_ws, size_t ws_size,
                              hipStream_t stream) {
  const float* h     = (const float*)d_in[0];
  // d_in[1] = distances: unused by the reference
  const int*   edges = (const int*)d_in[2];
  const float* nmask = (const float*)d_in[3];
  const float* emask = (const float*)d_in[4];
  const float* W1    = (const float*)d_in[5];
  const float* b1    = (const float*)d_in[6];
  const float* W2    = (const float*)d_in[7];
  const float* b2    = (const float*)d_in[8];
  const float* Wout  = (const float*)d_in[9];
  const float* bout  = (const float*)d_in[10];
  float* out = (float*)d_out;

  const int N = in_sizes[0] / DHID;
  const int E = in_sizes[4];
  size_t nd = (size_t)N * DHID;

  float* B0 = (float*)d_ws;   // current hyperbolic h
  float* B1 = B0 + nd;        // logmap / pre-GEMM
  float* B2 = B1 + nd;        // xlin
  float* B3 = B2 + nd;        // agg

  int rowBlocks  = (N + 7) / 8;
  int gemmBlocks = (N + 15) / 16;
  long long n4   = (long long)nd / 4;
  int zeroBlocks = (int)((n4 + 255) / 256);
  long long st   = (long long)E * 32;
  int scatBlocks = (int)((st + 255) / 256);
  int outBlocks  = ((N + 15) / 16 + 7) / 8;

  k_expmap0<<<rowBlocks, 256, 0, stream>>>(h, B0, N);

  for (int layer = 0; layer < 2; ++layer) {
    const float* W = layer ? W2 : W1;
    const float* b = layer ? b2 : b1;
    k_logmap0<<<rowBlocks, 256, 0, stream>>>(B0, B1, N);
    k_gemm128<<<gemmBlocks, 256, 0, stream>>>(B1, W, b, B2, N);
    k_zero<<<zeroBlocks, 256, 0, stream>>>((float4*)B3, n4);
    k_scatter<<<scatBlocks, 256, 0, stream>>>(B2, edges, emask, B3, E);
    k_relu_expmap<<<rowBlocks, 256, 0, stream>>>(B2, B3, nmask, B0, N);
  }

  k_logmap0<<<rowBlocks, 256, 0, stream>>>(B0, B1, N);
  k_gemm_out<<<outBlocks, 256, 0, stream>>>(B1, Wout, bout, out, N);
}